// PinPos_70214125355241
// MI455X (gfx1250) — compile-verified
//
#include <hip/hip_runtime.h>
#include <hip/hip_bf16.h>

// Pin position scatter-gather:
//   pin_x[i] = pos[node[i]]             + pin_offset_x[i]
//   pin_y[i] = pos[num_nodes + node[i]] + pin_offset_y[i]
//   out = [pin_x (num_pins), pin_y (num_pins)]
//
// Pure bandwidth-bound kernel. Streamed arrays use 128-bit non-temporal
// loads/stores (read-once / write-once) so the 16MB `pos` array stays
// resident in the 192MB L2 for the random gathers.

typedef float v4f __attribute__((ext_vector_type(4)));
typedef int   v4i __attribute__((ext_vector_type(4)));

__global__ __launch_bounds__(256)
void pinpos_gather_kernel(const float* __restrict__ pos,
                          const float* __restrict__ pin_offset_x,
                          const float* __restrict__ pin_offset_y,
                          const int*   __restrict__ pin2node,
                          float* __restrict__ out,
                          int num_pins, int num_nodes) {
    const int i4   = blockIdx.x * blockDim.x + threadIdx.x;  // vec4 index
    const int base = i4 * 4;
    const float* __restrict__ pos_y = pos + num_nodes;
    float* __restrict__ out_y = out + num_pins;

    if (base + 3 < num_pins) {
        // Read-once streams: non-temporal 128-bit loads.
        v4i n  = __builtin_nontemporal_load((const v4i*)pin2node     + i4);
        v4f ox = __builtin_nontemporal_load((const v4f*)pin_offset_x + i4);
        v4f oy = __builtin_nontemporal_load((const v4f*)pin_offset_y + i4);

        // Gathers: regular temporal loads — pos (16MB) should stay in L2.
        v4f px, py;
        px.x = pos[n.x];  px.y = pos[n.y];  px.z = pos[n.z];  px.w = pos[n.w];
        py.x = pos_y[n.x]; py.y = pos_y[n.y]; py.z = pos_y[n.z]; py.w = pos_y[n.w];

        px += ox;
        py += oy;

        // Write-once streams: non-temporal 128-bit stores.
        __builtin_nontemporal_store(px, (v4f*)out   + i4);
        __builtin_nontemporal_store(py, (v4f*)out_y + i4);
    } else if (base < num_pins) {
        // Scalar tail (not hit when num_pins % 4 == 0, kept for correctness).
        for (int i = base; i < num_pins; ++i) {
            int n = pin2node[i];
            out[i]   = pos[n]   + pin_offset_x[i];
            out_y[i] = pos_y[n] + pin_offset_y[i];
        }
    }
}

extern "C" void kernel_launch(void* const* d_in, const int* in_sizes, int n_in,
                              void* d_out, int out_size, void* d_ws, size_t ws_size,
                              hipStream_t stream) {
    // Inputs in setup_inputs() order:
    //   0: pos                      float32  [2*num_nodes]
    //   1: pin_offset_x             float32  [num_pins]
    //   2: pin_offset_y             float32  [num_pins]
    //   3: pin2node_map             int      [num_pins]
    //   4: flat_node2pin_map        (unused)
    //   5: flat_node2pin_start_map  (unused)
    const float* pos  = (const float*)d_in[0];
    const float* pox  = (const float*)d_in[1];
    const float* poy  = (const float*)d_in[2];
    const int*   p2n  = (const int*)d_in[3];
    float* out = (float*)d_out;

    const int num_nodes = in_sizes[0] / 2;
    const int num_pins  = in_sizes[1];

    const int threads = 256;                       // 8 waves per WGP-visit (wave32)
    const int num_vec = (num_pins + 3) / 4;        // 4 pins per thread
    const int blocks  = (num_vec + threads - 1) / threads;

    pinpos_gather_kernel<<<blocks, threads, 0, stream>>>(
        pos, pox, poy, p2n, out, num_pins, num_nodes);
}